// HiPPOEncoder_69389491634370
// MI455X (gfx1250) — compile-verified
//
#include <hip/hip_runtime.h>

// ---------------------------------------------------------------------------
// HiPPO encoder on MI455X (gfx1250), wave32 + V_WMMA_F32_16X16X4_F32.
//
// Recurrence x_l = Ab x_{l-1} + u_l*Bb re-blocked with stride T=32:
//   x_l = Ab^T x_{l-T} + sum_{j<T} u_{l-j} v_j ,   v_j = Ab^j Bb
// so 32 consecutive states form the columns of X_m (64x32) and
//   X_m = Ab^32 @ X_{m-1} + V @ U_m      (U_m[j,t] = u[m*T + t - j])
// -> 128 sequential (64x32x64 + 64x32x32) f32 GEMMs per head instead of
// 4096 GEMVs.  f32 WMMA keeps reference precision over the long scan.
//
// One workgroup (8 wave32) per head:
//  * LDS Gauss-Jordan inverse of I - dt/2 A (lower triangular, diag>=1,
//    pivot-free), then Ab, v_j, Ab^32 by squaring — all in-kernel.
//  * Ab^32 lower triangular -> wave row-tile mt skips k-chunks > 4*mt+3;
//    the bound goes through readfirstlane so the skip is a scalar branch
//    (no saveexec dance; EXEC stays all-ones for WMMA).
//  * X double-buffered (aug tail / dead Pbuf): 1 barrier per super-step.
//  * Two WMMA accumulators halve the dependent D->C chain per step.
//  * Coalesced float4 stores of the 256 MB output (~11 us @ 23.3 TB/s).
// ---------------------------------------------------------------------------

typedef float v2f __attribute__((ext_vector_type(2)));
typedef float v8f __attribute__((ext_vector_type(8)));

constexpr int kH = 256;
constexpr int kN = 64;
constexpr int kL = 4096;
constexpr int kT = 32;          // time columns per super-step
constexpr int kSup = kL / kT;   // 128 sequential super-steps

__device__ __forceinline__ void sq_gemm_lds(float* D, int ds, const float* S,
                                            int ss, int tid) {
  // D = S @ S, 64x64, both in LDS (disjoint element sets at every call site)
  __syncthreads();
  for (int idx = tid; idx < 64 * 64; idx += 256) {
    const int r = idx >> 6, c = idx & 63;
    float acc = 0.f;
#pragma unroll 8
    for (int k = 0; k < 64; ++k) acc += S[r * ss + k] * S[k * ss + c];
    D[r * ds + c] = acc;
  }
  __syncthreads();
}

__global__ __launch_bounds__(256) void hippo_scan_kernel(
    const float* __restrict__ u, const float* __restrict__ A,
    const float* __restrict__ Bc, const float* __restrict__ log_dt,
    float* __restrict__ out) {
  __shared__ __align__(16) float aug[64 * 128];  // GJ workspace; later u|X0
  __shared__ __align__(16) float Pbuf[64 * 64];  // Ab, Ab^32; later X1
  __shared__ __align__(16) float Vbuf[64 * kT];  // V[n*kT + j]
  __shared__ float fcol[64];
  __shared__ float pivs;

  const int tid = threadIdx.x;
  const int h = blockIdx.x;
  const float dt = log_dt[h];
  const float hdt = 0.5f * dt;
  const float* Ah = A + (size_t)h * kN * kN;

  // ---- build [ I - dt/2 A | I ] ----
  for (int idx = tid; idx < 64 * 128; idx += 256) {
    const int r = idx >> 7, c = idx & 127;
    aug[idx] = (c < 64) ? ((r == c ? 1.f : 0.f) - hdt * Ah[r * 64 + c])
                        : (((c - 64) == r) ? 1.f : 0.f);
  }
  __syncthreads();

  // ---- Gauss-Jordan (lower triangular, diag >= 1: pivot-free) ----
  for (int p = 0; p < 64; ++p) {
    if (tid == 0) pivs = aug[p * 128 + p];
    __syncthreads();
    const float inv = 1.0f / pivs;
    if (tid < 128) aug[p * 128 + tid] *= inv;
    __syncthreads();
    if (tid < 64) fcol[tid] = aug[tid * 128 + p];
    __syncthreads();
    for (int idx = tid; idx < 64 * 128; idx += 256) {
      const int r = idx >> 7, c = idx & 127;
      if (r != p) aug[idx] -= fcol[r] * aug[p * 128 + c];
    }
    __syncthreads();
  }
  // aug[:,64:128] = BL = inv(I - dt/2 A)

  // ---- Bb = dt * BL @ B  -> V[:,0] ----
  if (tid < 64) {
    float s = 0.f;
    for (int k = 0; k < 64; ++k) s += aug[tid * 128 + 64 + k] * Bc[h * 64 + k];
    Vbuf[tid * kT] = dt * s;
  }
  // ---- Ab = BL + dt/2 * (BL @ A)  -> Pbuf ----
  for (int idx = tid; idx < 64 * 64; idx += 256) {
    const int r = idx >> 6, c = idx & 63;
    float s = 0.f;
    for (int k = 0; k < 64; ++k) s += aug[r * 128 + 64 + k] * Ah[k * 64 + c];
    Pbuf[idx] = aug[r * 128 + 64 + c] + hdt * s;
  }
  __syncthreads();

  // ---- v_j = Ab v_{j-1}, j = 1..T-1 ----
  for (int j = 1; j < kT; ++j) {
    if (tid < 64) {
      float s = 0.f;
      for (int k = 0; k < 64; ++k)
        s += Pbuf[tid * 64 + k] * Vbuf[k * kT + j - 1];
      Vbuf[tid * kT + j] = s;
    }
    __syncthreads();
  }

  // ---- Ab^32 by squaring: Pbuf -> augL -> augR -> augL -> augR -> Pbuf ----
  sq_gemm_lds(aug, 128, Pbuf, 64, tid);       // Ab^2
  sq_gemm_lds(aug + 64, 128, aug, 128, tid);  // Ab^4  (BL dead)
  sq_gemm_lds(aug, 128, aug + 64, 128, tid);  // Ab^8
  sq_gemm_lds(aug + 64, 128, aug, 128, tid);  // Ab^16
  sq_gemm_lds(Pbuf, 64, aug + 64, 128, tid);  // Ab^32 (Ab dead)

  // ---- per-wave WMMA fragments (ISA 7.12.2 f32 layouts) ----
  const int lane = tid & 31;
  const int wave = tid >> 5;
  const int mt = wave & 3;        // state-row tile (M)
  const int nt = wave >> 2;       // time-col tile (N)
  const int hi = lane >> 4;       // lane half: K pair select / M offset
  const int lm = lane & 15;
  const int row = mt * 16 + lm;   // state row for A operands
  const int tcol = nt * 16 + lm;  // time column for B/C operands

  v2f afrag[16];
#pragma unroll
  for (int kc = 0; kc < 16; ++kc) {
    const int k0 = kc * 4 + 2 * hi;
    afrag[kc].x = Pbuf[row * 64 + k0];
    afrag[kc].y = Pbuf[row * 64 + k0 + 1];
  }
  v2f vfrag[8];
#pragma unroll
  for (int kc = 0; kc < 8; ++kc) {
    const int j0 = kc * 4 + 2 * hi;
    vfrag[kc].x = Vbuf[row * kT + j0];
    vfrag[kc].y = Vbuf[row * kT + j0 + 1];
  }

  // ---- stage zero-padded u; zero X0.  (X1 = Pbuf, written in iter 0) ----
  float* lds_u = aug;               // kT zeros then u[0..L)
  float* lds_X0 = aug + (kT + kL);  // X[t*64 + n], column-major 64 x kT
  float* lds_X1 = Pbuf;             // dead after fragment preload
  for (int i = tid; i < kT; i += 256) lds_u[i] = 0.f;
  for (int i = tid; i < kL / 4; i += 256)
    ((float4*)(lds_u + kT))[i] = ((const float4*)u)[i];
  for (int i = tid; i < kN * kT; i += 256) lds_X0[i] = 0.f;
  __syncthreads();  // fragments preloaded + u/X0 staged before scan

  // Ab^32 lower triangular: wave row-tile mt needs k-chunks < 4*mt+4.
  // readfirstlane -> SGPR bound -> scalar branches, EXEC untouched.
  const int kcmax = __builtin_amdgcn_readfirstlane(4 * mt + 4);
  const size_t outhead = (size_t)h * kN;

  for (int m = 0; m < kSup; ++m) {
    float* __restrict__ Xr = (m & 1) ? lds_X1 : lds_X0;  // read buffer
    float* __restrict__ Xw = (m & 1) ? lds_X0 : lds_X1;  // write buffer

    v8f c0 = {0.f, 0.f, 0.f, 0.f, 0.f, 0.f, 0.f, 0.f};
    v8f c1 = {0.f, 0.f, 0.f, 0.f, 0.f, 0.f, 0.f, 0.f};
    // X_new(tile) += Ab^32(row-tile) @ X  (two accumulators break the chain)
#pragma unroll
    for (int kc = 0; kc < 16; kc += 2) {
      if (kc < kcmax) {
        const int k0 = kc * 4 + 2 * hi;
        v2f b;
        b.x = Xr[tcol * 64 + k0];
        b.y = Xr[tcol * 64 + k0 + 1];
        c0 = __builtin_amdgcn_wmma_f32_16x16x4_f32(false, afrag[kc], false, b,
                                                   (short)0, c0, false, false);
      }
      if (kc + 1 < kcmax) {
        const int k0 = (kc + 1) * 4 + 2 * hi;
        v2f b;
        b.x = Xr[tcol * 64 + k0];
        b.y = Xr[tcol * 64 + k0 + 1];
        c1 = __builtin_amdgcn_wmma_f32_16x16x4_f32(false, afrag[kc + 1], false,
                                                   b, (short)0, c1, false,
                                                   false);
      }
    }
    // X_new(tile) += V(row-tile) @ U_m,  U_m[j,t] = u[m*T + t - j]
    const int ubase = kT + m * kT + tcol;
#pragma unroll
    for (int kc = 0; kc < 8; ++kc) {
      const int j0 = kc * 4 + 2 * hi;
      v2f b;
      b.x = lds_u[ubase - j0];
      b.y = lds_u[ubase - j0 - 1];
      if (kc & 1)
        c1 = __builtin_amdgcn_wmma_f32_16x16x4_f32(false, vfrag[kc], false, b,
                                                   (short)0, c1, false, false);
      else
        c0 = __builtin_amdgcn_wmma_f32_16x16x4_f32(false, vfrag[kc], false, b,
                                                   (short)0, c0, false, false);
    }
    const v8f cc = c0 + c1;
    {
      const int nbase = mt * 16 + hi * 8;  // C/D: VGPR r -> rows r, r+8
      float4 lo = make_float4(cc[0], cc[1], cc[2], cc[3]);
      float4 hi4 = make_float4(cc[4], cc[5], cc[6], cc[7]);
      *(float4*)&Xw[tcol * 64 + nbase] = lo;
      *(float4*)&Xw[tcol * 64 + nbase + 4] = hi4;
    }
    __syncthreads();  // X_m visible; prior iter's LDS reads drained
    // coalesced stores: out[(m*T+t)*H*N + h*N + n]
    for (int q = tid; q < (kN * kT) / 4; q += 256) {
      const int t = q >> 4;
      const int n4 = (q & 15) * 4;
      float4 val = *(const float4*)&Xw[t * 64 + n4];
      *(float4*)&out[(size_t)(m * kT + t) * (kH * kN) + outhead + n4] = val;
    }
  }
}

extern "C" void kernel_launch(void* const* d_in, const int* in_sizes, int n_in,
                              void* d_out, int out_size, void* d_ws,
                              size_t ws_size, hipStream_t stream) {
  (void)in_sizes; (void)n_in; (void)out_size; (void)d_ws; (void)ws_size;
  const float* u = (const float*)d_in[0];       // L
  const float* A = (const float*)d_in[1];       // H*N*N
  const float* B = (const float*)d_in[2];       // H*N
  const float* log_dt = (const float*)d_in[3];  // H
  float* out = (float*)d_out;                   // L*H*N f32
  hippo_scan_kernel<<<dim3(kH), dim3(256), 0, stream>>>(u, A, B, log_dt, out);
}